// MultiHeadAttentionLayer_14508399526689
// MI455X (gfx1250) — compile-verified
//
#include <hip/hip_runtime.h>

typedef __attribute__((ext_vector_type(2))) float v2f;
typedef __attribute__((ext_vector_type(8))) float v8f;

#define FDIM 128

// ---------------------------------------------------------------------------
// Zero helper (graph-replay safe re-init of accumulators)
// ---------------------------------------------------------------------------
__global__ __launch_bounds__(256) void zero_f32(float* __restrict__ p, long long n) {
  long long i = (long long)blockIdx.x * 256 + threadIdx.x;
  if (i < n) p[i] = 0.0f;
}

// ---------------------------------------------------------------------------
// C[M,128] = A[M,128] @ B[128,128], f32, via V_WMMA_F32_16X16X4_F32.
// One block = 16 rows of A (staged in LDS); 8 waves, each owns a 16-col tile.
// M is a multiple of 16 (50000 and 400000 both are).
// ---------------------------------------------------------------------------
__global__ __launch_bounds__(256) void gemm16x128_f32_wmma(
    const float* __restrict__ A, const float* __restrict__ B,
    float* __restrict__ C)
{
  __shared__ float sA[16 * FDIM];  // 8 KB

  const int tid = threadIdx.x;
  const size_t rowbase = (size_t)blockIdx.x * 16;

  // Cooperative load of the 16x128 A tile: 512 float4, 2 per thread.
  {
    const float4* Ag = (const float4*)(A + rowbase * FDIM);
    float4* As = (float4*)sA;
    As[tid]       = Ag[tid];
    As[tid + 256] = Ag[tid + 256];
  }
  __syncthreads();

  const int wave    = tid >> 5;        // 0..7 -> 16-wide column tile
  const int lane    = tid & 31;
  const int colbase = wave * 16;
  const int lhalf   = lane >> 4;       // 0: lanes 0-15, 1: lanes 16-31
  const int lmod    = lane & 15;

  v8f acc = {0.f, 0.f, 0.f, 0.f, 0.f, 0.f, 0.f, 0.f};

  // A frag 16x4: lane (l<16): row=lmod, K=k0+{0,1}; l>=16: row=lmod, K=k0+{2,3}
  // B frag 4x16: VGPR j, lane l: K=k0+j+2*lhalf, N=lmod
  #pragma unroll 4
  for (int k0 = 0; k0 < FDIM; k0 += 4) {
    v2f a = *(const v2f*)(&sA[lmod * FDIM + k0 + 2 * lhalf]);
    const float* Bp = B + (size_t)(k0 + 2 * lhalf) * FDIM + colbase + lmod;
    v2f b;
    b.x = Bp[0];
    b.y = Bp[FDIM];
    acc = __builtin_amdgcn_wmma_f32_16x16x4_f32(
        /*neg_a=*/false, a, /*neg_b=*/false, b,
        /*c_mod=*/(short)0, acc, /*reuse_a=*/false, /*reuse_b=*/false);
  }

  // C/D 16x16 layout: VGPR r -> row r (lanes 0-15) or r+8 (lanes 16-31), col lmod
  float* Cw = C + rowbase * FDIM + colbase;
  #pragma unroll
  for (int r = 0; r < 8; ++r)
    Cw[(size_t)(r + 8 * lhalf) * FDIM + lmod] = acc[r];
}

// ---------------------------------------------------------------------------
// Edge kernel: one thread per (edge, h, d). e_proj holds proj_e on entry and
// is overwritten with the score (= e_out). Scatter via f32 global atomics.
// ---------------------------------------------------------------------------
__global__ __launch_bounds__(256) void edge_kernel(
    const float* __restrict__ Q, const float* __restrict__ K,
    const float* __restrict__ V,
    float* __restrict__ e_proj,          // [E,128]: in proj_e, out score
    const int* __restrict__ src, const int* __restrict__ dst,
    float* __restrict__ wV,              // [N,128] accumulator (h_out region)
    float* __restrict__ z,               // [N,8] accumulator
    long long E_total)
{
  long long t = (long long)blockIdx.x * 256 + threadIdx.x;
  if (t >= E_total * FDIM) return;

  const int e = (int)(t >> 7);
  const int f = (int)(t & (FDIM - 1));   // h*16 + d
  const int h = f >> 4;

  const int s  = src[e];
  const int dn = dst[e];

  const float k  = K[(size_t)s * FDIM + f];
  const float q  = Q[(size_t)dn * FDIM + f];
  const float pe = e_proj[t];

  float sc = k * q * 0.25f;                       // / sqrt(D), D=16
  sc = fminf(5.0f, fmaxf(-5.0f, sc));
  sc *= pe;
  e_proj[t] = sc;                                 // e_out

  // Per-head sum over d (16 lanes). Wave32 holds two heads; xor masks 1..8
  // stay within each 16-lane half.
  float sum = sc;
  #pragma unroll
  for (int m = 1; m < 16; m <<= 1)
    sum += __shfl_xor(sum, m, 32);
  sum = fminf(5.0f, fmaxf(-5.0f, sum));
  const float attn = __expf(sum);

  const float v = V[(size_t)s * FDIM + f];
  atomicAdd(&wV[(size_t)dn * FDIM + f], v * attn);
  if ((f & 15) == 0)
    atomicAdd(&z[(size_t)dn * 8 + h], attn);
}

// ---------------------------------------------------------------------------
// h_out = wV / (z + 1e-6), in place over the h_out region of d_out.
// ---------------------------------------------------------------------------
__global__ __launch_bounds__(256) void finalize_kernel(
    float* __restrict__ h_out, const float* __restrict__ z, long long total)
{
  long long t = (long long)blockIdx.x * 256 + threadIdx.x;
  if (t >= total) return;
  const int node = (int)(t >> 7);
  const int h = (int)((t >> 4) & 7);
  h_out[t] = h_out[t] / (z[(size_t)node * 8 + h] + 1e-6f);
}

// ---------------------------------------------------------------------------
extern "C" void kernel_launch(void* const* d_in, const int* in_sizes, int n_in,
                              void* d_out, int out_size, void* d_ws, size_t ws_size,
                              hipStream_t stream)
{
  const float* node_feats = (const float*)d_in[0];
  const float* edge_feats = (const float*)d_in[1];
  const float* Wq = (const float*)d_in[2];
  const float* Wk = (const float*)d_in[3];
  const float* Wv = (const float*)d_in[4];
  const float* We = (const float*)d_in[5];
  const int*   src = (const int*)d_in[6];
  const int*   dst = (const int*)d_in[7];

  const long long N = (long long)in_sizes[0] / FDIM;
  const long long E = (long long)in_sizes[6];

  float* h_out = (float*)d_out;                  // [N,128] (wV accum, then normalized)
  float* e_out = (float*)d_out + N * FDIM;       // [E,128] (proj_e, then score)

  float* Q  = (float*)d_ws;                      // [N,128]
  float* Kp = Q  + N * FDIM;                     // [N,128]
  float* Vp = Kp + N * FDIM;                     // [N,128]
  float* z  = Vp + N * FDIM;                     // [N,8]

  // 1) zero accumulators (must happen every call: graph replays)
  {
    long long n = N * FDIM;
    zero_f32<<<(unsigned)((n + 255) / 256), 256, 0, stream>>>(h_out, n);
    long long nz = N * 8;
    zero_f32<<<(unsigned)((nz + 255) / 256), 256, 0, stream>>>(z, nz);
  }

  // 2) projections (f32 WMMA GEMMs)
  gemm16x128_f32_wmma<<<(unsigned)(N / 16), 256, 0, stream>>>(node_feats, Wq, Q);
  gemm16x128_f32_wmma<<<(unsigned)(N / 16), 256, 0, stream>>>(node_feats, Wk, Kp);
  gemm16x128_f32_wmma<<<(unsigned)(N / 16), 256, 0, stream>>>(node_feats, Wv, Vp);
  gemm16x128_f32_wmma<<<(unsigned)(E / 16), 256, 0, stream>>>(edge_feats, We, e_out);

  // 3) edge scores + scatter
  {
    long long tot = E * FDIM;
    edge_kernel<<<(unsigned)((tot + 255) / 256), 256, 0, stream>>>(
        Q, Kp, Vp, e_out, src, dst, h_out, z, E);
  }

  // 4) normalize
  {
    long long hn = N * FDIM;
    finalize_kernel<<<(unsigned)((hn + 255) / 256), 256, 0, stream>>>(h_out, z, hn);
  }
}